// LDNCellCompact_9732395893055
// MI455X (gfx1250) — compile-verified
//
#include <hip/hip_runtime.h>

// ---------------------------------------------------------------------------
// LDN cell on MI455X (gfx1250, wave32, WMMA).
//   Ad = expm(A)            (scaling-and-squaring Taylor, s=18, K=24 terms)
//   Bd = phi1(A) @ B        (phi1 doubling: b <- 0.5*(E b + b) before E <- E^2)
//   new_state = Ad @ state + Bd x^T
// Outputs: [new_state.T.flatten() | new_state.flatten()]
// ---------------------------------------------------------------------------

typedef float v2f __attribute__((ext_vector_type(2)));
typedef float v8f __attribute__((ext_vector_type(8)));

#define QDIM 512
#define NSEQ 65536

// CDNA5 async memory->LDS path (ASYNCcnt-tracked), guarded so compilation
// never fails if this toolchain lacks the builtins.
#if defined(__AMDGCN__) && \
    __has_builtin(__builtin_amdgcn_global_load_async_to_lds_b128) && \
    __has_builtin(__builtin_amdgcn_s_wait_asynccnt)
#define LDN_ASYNC_LDS 1
// Exact pointee type from the hipcc diagnostic:
//   '__attribute__((__vector_size__(4 * sizeof(int)))) int __device__ *'
typedef int ldn_v4i __attribute__((vector_size(4 * sizeof(int))));
typedef __attribute__((address_space(1))) ldn_v4i* ldn_gv4i_ptr;  // prints as __device__
typedef __attribute__((address_space(3))) ldn_v4i* ldn_lv4i_ptr;  // prints as __shared__
#else
#define LDN_ASYNC_LDS 0
#endif

static __device__ __forceinline__ v8f wmma_f32(v2f a, v2f b, v8f c) {
    // 8 args: (neg_a, A, neg_b, B, c_mod, C, reuse_a, reuse_b)
    return __builtin_amdgcn_wmma_f32_16x16x4_f32(false, a, false, b,
                                                 (short)0, c, false, false);
}

// ---------------- small setup kernels --------------------------------------

// T = A * 2^-s ; E = I + T ; M1 = T   (Taylor term_1)
__global__ __launch_bounds__(256)
void ldn_init_TE(const float* __restrict__ A, float* __restrict__ T,
                 float* __restrict__ E, float* __restrict__ M1, float scale) {
    int idx = blockIdx.x * 256 + threadIdx.x;
    if (idx >= QDIM * QDIM) return;
    float t = A[idx] * scale;
    T[idx]  = t;
    M1[idx] = t;
    int i = idx >> 9, j = idx & (QDIM - 1);
    E[idx] = t + (i == j ? 1.0f : 0.0f);
}

// b = B ; v1 = B   (phi1 Taylor: term_0 = B / 1!)
__global__ __launch_bounds__(256)
void ldn_init_vec(const float* __restrict__ B, float* __restrict__ b,
                  float* __restrict__ v1) {
    int i = blockIdx.x * 256 + threadIdx.x;
    if (i < QDIM) { b[i] = B[i]; v1[i] = B[i]; }
}

// v2 = (T @ v1) * coef ; b += v2      (phi1 Taylor term recurrence)
__global__ __launch_bounds__(256)
void ldn_matvec_acc(const float* __restrict__ T, const float* __restrict__ v1,
                    float* __restrict__ v2, float* __restrict__ b, float coef) {
    int row = blockIdx.x * 256 + threadIdx.x;
    if (row >= QDIM) return;
    float s = 0.0f;
    const float* tr = T + (size_t)row * QDIM;
    for (int k = 0; k < QDIM; ++k) s = fmaf(tr[k], v1[k], s);
    s *= coef;
    v2[row] = s;
    b[row] += s;
}

// bout = 0.5 * (E @ bin + bin)        (phi1 doubling step, uses E BEFORE squaring)
__global__ __launch_bounds__(256)
void ldn_phi_step(const float* __restrict__ E, const float* __restrict__ bin,
                  float* __restrict__ bout) {
    int row = blockIdx.x * 256 + threadIdx.x;
    if (row >= QDIM) return;
    float s = bin[row];
    const float* er = E + (size_t)row * QDIM;
    for (int k = 0; k < QDIM; ++k) s = fmaf(er[k], bin[k], s);
    bout[row] = 0.5f * s;
}

// ---------------- 512x512x512 WMMA matmul ----------------------------------
// D = (Am @ Bm) * scale ; if ACC, Eacc += D.  One wave per 16x16 tile.
template <bool ACC>
__global__ __launch_bounds__(32)
void ldn_wmma_mm512(const float* __restrict__ Am, const float* __restrict__ Bm,
                    float* __restrict__ Dm, float* __restrict__ Eacc,
                    float scale) {
    const int lane = threadIdx.x;
    const int half = lane >> 4;        // selects K pair (A/B), M half (C)
    const int l15  = lane & 15;
    const int m0 = blockIdx.x * 16;
    const int n0 = blockIdx.y * 16;

    v8f c = {};
    for (int k = 0; k < QDIM; k += 4) {
        const int ka = k + half * 2;
        // A 16x4: lane m=l15 holds K=ka, ka+1 (contiguous -> float2 load)
        v2f a = *(const v2f*)&Am[(size_t)(m0 + l15) * QDIM + ka];
        // B 4x16: lane n=l15 holds K=ka, ka+1 (rows of Bm, stride QDIM)
        v2f b;
        b.x = Bm[(size_t)ka * QDIM + n0 + l15];
        b.y = Bm[(size_t)(ka + 1) * QDIM + n0 + l15];
        c = wmma_f32(a, b, c);
    }
    // C/D: VGPR r = row m0 + r + half*8, col n0 + l15
#pragma unroll
    for (int r = 0; r < 8; ++r) {
        int row = m0 + r + half * 8;
        float v = c[r] * scale;
        size_t o = (size_t)row * QDIM + n0 + l15;
        Dm[o] = v;
        if (ACC) Eacc[o] += v;
    }
}

// ---------------- big fused GEMM + rank-1 + dual-layout store --------------
// Per block: one 16-column tile of state (staged once into 32KB LDS -> state
// read from HBM exactly once), 8 waves x 4 accumulators cover all 512 rows.
__global__ __launch_bounds__(256)
void ldn_final(const float* __restrict__ E, const float* __restrict__ state,
               const float* __restrict__ bvec, const float* __restrict__ x,
               float* __restrict__ outT, float* __restrict__ outR) {
    __shared__ float sTile[QDIM * 16];   // [k][n] : 32 KB
    const int n0  = blockIdx.x * 16;
    const int tid = threadIdx.x;

    // Hint: keep the (L2-resident) Ad matrix warm.
    __builtin_prefetch(&E[(size_t)tid * 1024], 0, 2);

    // Stage state[:, n0:n0+16] into LDS in 16B chunks (2048 float4s / block).
#if LDN_ASYNC_LDS
#pragma unroll
    for (int it = 0; it < 8; ++it) {
        int lin = it * 256 + tid;           // 0..2047
        int k = lin >> 2, q4 = lin & 3;
        __builtin_amdgcn_global_load_async_to_lds_b128(
            (ldn_gv4i_ptr)&state[(size_t)k * NSEQ + n0 + q4 * 4],
            (ldn_lv4i_ptr)&sTile[k * 16 + q4 * 4],
            0, 0);
    }
    __builtin_amdgcn_s_wait_asynccnt(0);
#else
#pragma unroll
    for (int it = 0; it < 8; ++it) {
        int lin = it * 256 + tid;           // 0..2047
        int k = lin >> 2, q4 = lin & 3;
        float4 v = *(const float4*)&state[(size_t)k * NSEQ + n0 + q4 * 4];
        *(float4*)&sTile[k * 16 + q4 * 4] = v;
    }
#endif
    __syncthreads();

    const int wave = tid >> 5;
    const int lane = tid & 31;
    const int half = lane >> 4;
    const int l15  = lane & 15;
    const int mB   = wave * 64;             // this wave's 64-row strip

    v8f c0 = {}, c1 = {}, c2 = {}, c3 = {};
    for (int k = 0; k < QDIM; k += 4) {
        const int kb = k + half * 2;
        v2f b;
        b.x = sTile[kb * 16 + l15];
        b.y = sTile[(kb + 1) * 16 + l15];
        v2f a0 = *(const v2f*)&E[(size_t)(mB +  0 + l15) * QDIM + kb];
        v2f a1 = *(const v2f*)&E[(size_t)(mB + 16 + l15) * QDIM + kb];
        v2f a2 = *(const v2f*)&E[(size_t)(mB + 32 + l15) * QDIM + kb];
        v2f a3 = *(const v2f*)&E[(size_t)(mB + 48 + l15) * QDIM + kb];
        c0 = wmma_f32(a0, b, c0);
        c1 = wmma_f32(a1, b, c1);
        c2 = wmma_f32(a2, b, c2);
        c3 = wmma_f32(a3, b, c3);
    }

    const int n   = n0 + l15;
    const float xn = x[n];
    v8f* accs[4] = { &c0, &c1, &c2, &c3 };
#pragma unroll
    for (int sub = 0; sub < 4; ++sub) {
        v8f c = *accs[sub];
        const int mSub = mB + sub * 16 + half * 8;   // 8 consecutive rows
#pragma unroll
        for (int r = 0; r < 8; ++r)
            c[r] = fmaf(bvec[mSub + r], xn, c[r]);
        // row-major half of output: outR[q*NSEQ + n]
#pragma unroll
        for (int r = 0; r < 8; ++r)
            outR[(size_t)(mSub + r) * NSEQ + n] = c[r];
        // transposed half: outT[n*QDIM + q] -> 8 contiguous floats per lane
        float4 lo = { c[0], c[1], c[2], c[3] };
        float4 hi = { c[4], c[5], c[6], c[7] };
        *(float4*)&outT[(size_t)n * QDIM + mSub]     = lo;
        *(float4*)&outT[(size_t)n * QDIM + mSub + 4] = hi;
    }
}

// ---------------------------------------------------------------------------

extern "C" void kernel_launch(void* const* d_in, const int* in_sizes, int n_in,
                              void* d_out, int out_size, void* d_ws, size_t ws_size,
                              hipStream_t stream) {
    (void)in_sizes; (void)n_in; (void)out_size; (void)ws_size;
    const float* x     = (const float*)d_in[0];   // [65536]
    const float* state = (const float*)d_in[1];   // [512, 65536]
    const float* A     = (const float*)d_in[2];   // [512, 512]
    const float* B     = (const float*)d_in[3];   // [512]
    float* out = (float*)d_out;
    float* ws  = (float*)d_ws;

    const size_t QQ = (size_t)QDIM * QDIM;
    float* T  = ws;
    float* E  = ws + QQ;
    float* M1 = ws + 2 * QQ;
    float* M2 = ws + 3 * QQ;
    float* bv = ws + 4 * QQ;
    float* v1 = bv + QDIM;
    float* v2 = v1 + QDIM;

    const int S = 18;                  // ||A||inf ~ 2^17  ->  ||T|| ~ 0.5
    const int K = 24;                  // Taylor terms (0.5^24/24! << eps)
    const float scale = 1.0f / 262144.0f;   // 2^-18

    ldn_init_TE<<<(QDIM * QDIM) / 256, 256, 0, stream>>>(A, T, E, M1, scale);
    ldn_init_vec<<<2, 256, 0, stream>>>(B, bv, v1);

    dim3 g32(QDIM / 16, QDIM / 16);
    // E = sum_{k<=K} T^k/k!   (term_k = term_{k-1} @ T / k, accumulated into E)
    for (int k = 2; k <= K; ++k) {
        ldn_wmma_mm512<true><<<g32, 32, 0, stream>>>(M1, T, M2, E, 1.0f / (float)k);
        float* tmp = M1; M1 = M2; M2 = tmp;
    }
    // b = phi1(T) @ B  (vector Taylor: term_k = T term_{k-1} / (k+1))
    for (int k = 1; k <= K; ++k) {
        ldn_matvec_acc<<<2, 256, 0, stream>>>(T, v1, v2, bv, 1.0f / (float)(k + 1));
        float* tmp = v1; v1 = v2; v2 = tmp;
    }
    // Squaring phase: b <- 0.5 (E b + b) using E_i, then E <- E^2.
    for (int i = 0; i < S; ++i) {
        ldn_phi_step<<<2, 256, 0, stream>>>(E, bv, v1);
        float* tv = bv; bv = v1; v1 = tv;
        ldn_wmma_mm512<false><<<g32, 32, 0, stream>>>(E, E, M2, nullptr, 1.0f);
        float* tm = E; E = M2; M2 = tm;
    }

    float* outT = out;                               // new_state.T.flatten()
    float* outR = out + (size_t)QDIM * NSEQ;         // new_state.flatten()
    ldn_final<<<NSEQ / 16, 256, 0, stream>>>(E, state, bv, x, outT, outR);
}